// UCECollisionEntropyLoss_15444702397060
// MI455X (gfx1250) — compile-verified
//
#include <hip/hip_runtime.h>
#include <stdint.h>

#define UCE_NBINS 15

// ---------- helpers ----------

// Order-preserving float -> uint mapping (works for negative H2 too)
__device__ __forceinline__ unsigned f2mono(float f) {
    unsigned u = __float_as_uint(f);
    return (u & 0x80000000u) ? ~u : (u | 0x80000000u);
}
__device__ __forceinline__ float mono2f(unsigned m) {
    unsigned u = (m & 0x80000000u) ? (m & 0x7fffffffu) : ~m;
    return __uint_as_float(u);
}

// Per-row collision entropy + argmax over 16 classes (softmax is monotone,
// so argmax(probs) == argmax(logits); strict '>' keeps the first max like jnp.argmax)
__device__ __forceinline__ void row_h2_arg(const float* v, float& H2, int& arg) {
    float m = v[0]; arg = 0;
#pragma unroll
    for (int j = 1; j < 16; ++j) { if (v[j] > m) { m = v[j]; arg = j; } }
    float s1 = 0.f, s2 = 0.f;
#pragma unroll
    for (int j = 0; j < 16; ++j) {
        float e = __expf(v[j] - m);
        s1 += e;
        s2 += e * e;                       // sum exp(2(l-m))
    }
    float sp = s2 / (s1 * s1);             // sum p^2
    H2 = -__log2f(sp + 1e-12f);
}

// Issue 4 async b128 loads: one 64B row -> this lane's LDS slot.
// GVS mode: 64-bit SGPR base + 32-bit per-lane VGPR byte offset; INST_OFFSET
// applies to BOTH the global and LDS address per the ISA pseudocode.
__device__ __forceinline__ void async_row_load(unsigned ldsoff, unsigned goff,
                                               unsigned long long base) {
    asm volatile(
        "s_wait_dscnt 0x0\n\t"  // prior ds reads of this buffer fully drained
        "global_load_async_to_lds_b128 %0, %1, %2 offset:0\n\t"
        "global_load_async_to_lds_b128 %0, %1, %2 offset:16\n\t"
        "global_load_async_to_lds_b128 %0, %1, %2 offset:32\n\t"
        "global_load_async_to_lds_b128 %0, %1, %2 offset:48"
        :: "v"(ldsoff), "v"(goff), "s"(base) : "memory");
}

// ---------- kernel 0: init accumulators ----------
__global__ void uce_init(unsigned* __restrict__ minmax,
                         unsigned* __restrict__ gcount,
                         float* __restrict__ gsh,
                         float* __restrict__ gse) {
    int t = threadIdx.x;
    if (t == 0) { minmax[0] = 0xFFFFFFFFu; minmax[1] = 0u; }
    if (t < UCE_NBINS) { gcount[t] = 0u; gsh[t] = 0.f; gse[t] = 0.f; }
}

// ---------- kernel 1: stream logits via async->LDS, compute H2/err, min/max ----------
__global__ __launch_bounds__(256) void uce_pass1(
    const float* __restrict__ logits, const int* __restrict__ labels,
    unsigned* __restrict__ minmax, float* __restrict__ h2out,
    unsigned char* __restrict__ errout, int B, int store) {
    __shared__ float4 stage[2][256 * 4];   // ping-pong, 64B per lane per buffer (32 KB)

    const int tid = threadIdx.x;
    const int gsize = gridDim.x * blockDim.x;
    int i = blockIdx.x * blockDim.x + tid;

    unsigned long long base = (unsigned long long)logits;
    unsigned ldsA = (unsigned)(size_t)&stage[0][tid * 4];
    unsigned ldsB = (unsigned)(size_t)&stage[1][tid * 4];

    unsigned lmin = 0xFFFFFFFFu, lmax = 0u;

    if (i < B) async_row_load(ldsA, (unsigned)i * 64u, base);

    int buf = 0;
    for (; i < B; i += gsize) {
        int nxt = i + gsize;
        if (nxt < B) {
            async_row_load(buf ? ldsA : ldsB, (unsigned)nxt * 64u, base);
            // 8 async ops outstanding; the oldest 4 (current buffer) complete in order
            asm volatile("s_wait_asynccnt 0x4" ::: "memory");
        } else {
            asm volatile("s_wait_asynccnt 0x0" ::: "memory");
        }

        const float4* rp = buf ? &stage[1][tid * 4] : &stage[0][tid * 4];
        float4 q0 = rp[0], q1 = rp[1], q2 = rp[2], q3 = rp[3];
        float v[16] = {q0.x, q0.y, q0.z, q0.w, q1.x, q1.y, q1.z, q1.w,
                       q2.x, q2.y, q2.z, q2.w, q3.x, q3.y, q3.z, q3.w};

        float H2; int arg;
        row_h2_arg(v, H2, arg);

        if (store) {
            h2out[i] = H2;
            errout[i] = (arg != labels[i]) ? (unsigned char)1 : (unsigned char)0;
        }
        unsigned mh = f2mono(H2);
        lmin = (lmin < mh) ? lmin : mh;
        lmax = (lmax > mh) ? lmax : mh;
        buf ^= 1;
    }

    // wave32 min/max reduction, then one atomic pair per wave
#pragma unroll
    for (int off = 16; off > 0; off >>= 1) {
        unsigned a = __shfl_xor(lmin, off, 32); lmin = (lmin < a) ? lmin : a;
        unsigned b = __shfl_xor(lmax, off, 32); lmax = (lmax > b) ? lmax : b;
    }
    if ((tid & 31) == 0) {
        atomicMin(&minmax[0], lmin);
        atomicMax(&minmax[1], lmax);
    }
}

// ---------- kernel 2: bin + segment sums (LDS histogram -> global atomics) ----------
__global__ __launch_bounds__(256) void uce_pass2(
    const float* __restrict__ h2in, const unsigned char* __restrict__ errin,
    const float* __restrict__ logits, const int* __restrict__ labels,
    const unsigned* __restrict__ minmax,
    unsigned* __restrict__ gcount, float* __restrict__ gsh, float* __restrict__ gse,
    int B, int fromWs) {
    __shared__ unsigned sc[UCE_NBINS];
    __shared__ float sh[UCE_NBINS];
    __shared__ float se[UCE_NBINS];

    int tid = threadIdx.x;
    if (tid < UCE_NBINS) { sc[tid] = 0u; sh[tid] = 0.f; se[tid] = 0.f; }
    __syncthreads();

    float mn = mono2f(minmax[0]);
    float mx = mono2f(minmax[1]);
    float step = (mx - mn) / (float)UCE_NBINS;
    float edges[UCE_NBINS + 1];
#pragma unroll
    for (int j = 0; j <= UCE_NBINS; ++j) edges[j] = mn + step * (float)j;
    edges[UCE_NBINS] = mx + 1e-6f;   // linspace endpoint is exact, then += 1e-6

    int gsize = gridDim.x * blockDim.x;
    for (int i = blockIdx.x * blockDim.x + tid; i < B; i += gsize) {
        float H2, err;
        if (fromWs) {
            H2 = h2in[i];
            err = (float)errin[i];
        } else {
            const float4* rp = (const float4*)(logits + (size_t)i * 16);
            float4 q0 = rp[0], q1 = rp[1], q2 = rp[2], q3 = rp[3];
            float v[16] = {q0.x, q0.y, q0.z, q0.w, q1.x, q1.y, q1.z, q1.w,
                           q2.x, q2.y, q2.z, q2.w, q3.x, q3.y, q3.z, q3.w};
            int arg; row_h2_arg(v, H2, arg);
            err = (arg != labels[i]) ? 1.f : 0.f;
        }
        // searchsorted(edges, H2, 'left') == count of edges strictly < H2
        int c = 0;
#pragma unroll
        for (int j = 0; j <= UCE_NBINS; ++j) c += (edges[j] < H2) ? 1 : 0;
        int b = c - 1;                 // H2==min -> -1 -> dropped (matches reference)
        if (b >= 0 && b < UCE_NBINS) {
            atomicAdd(&sc[b], 1u);
            atomicAdd(&sh[b], H2);
            atomicAdd(&se[b], err);
        }
    }
    __syncthreads();
    if (tid < UCE_NBINS) {
        if (sc[tid]) atomicAdd(&gcount[tid], sc[tid]);
        if (sh[tid] != 0.f) atomicAdd(&gsh[tid], sh[tid]);
        if (se[tid] != 0.f) atomicAdd(&gse[tid], se[tid]);
    }
}

// ---------- kernel 3: finalize (one wave) ----------
__global__ void uce_final(const unsigned* __restrict__ gcount,
                          const float* __restrict__ gsh,
                          const float* __restrict__ gse,
                          float* __restrict__ out, int B) {
    int i = threadIdx.x;
    float term = 0.f, eb = 0.f, hb = 0.f;
    if (i < UCE_NBINS) {
        float c = (float)gcount[i];
        bool ne = c > 0.f;
        float den = ne ? c : 1.f;
        hb = ne ? gsh[i] / den : 0.f;
        eb = ne ? gse[i] / den : 0.f;
        // _risk_from_H2
        float inner = 2.f * exp2f(-hb) - 1.f;
        inner = inner > 0.f ? inner : 0.f;
        float s = inner > 0.f ? sqrtf(inner) : 0.f;
        float rr = 0.5f * (1.f - s);
        float prop = c / (float)B;
        term = ne ? fabsf(eb - rr) * prop : 0.f;
    }
#pragma unroll
    for (int off = 16; off > 0; off >>= 1) term += __shfl_xor(term, off, 32);
    if (i == 0) out[0] = term;                       // uce
    if (i < UCE_NBINS) {
        out[1 + i] = eb;                             // err_bar
        out[1 + UCE_NBINS + i] = hb;                 // H2_bar
    }
}

// ---------- host ----------
extern "C" void kernel_launch(void* const* d_in, const int* in_sizes, int n_in,
                              void* d_out, int out_size, void* d_ws, size_t ws_size,
                              hipStream_t stream) {
    (void)n_in; (void)out_size;
    const float* logits = (const float*)d_in[0];
    const int* labels   = (const int*)d_in[1];
    int B = in_sizes[1];                    // rows; in_sizes[0] == B*16
    float* out = (float*)d_out;

    unsigned char* ws = (unsigned char*)d_ws;
    unsigned* minmax = (unsigned*)(ws + 0);        // 2 u32
    unsigned* gcount = (unsigned*)(ws + 64);       // 15 u32
    float*    gsh    = (float*)(ws + 128);         // 15 f32
    float*    gse    = (float*)(ws + 192);         // 15 f32
    float*         h2arr  = (float*)(ws + 256);                     // B f32
    unsigned char* errarr = (unsigned char*)(ws + 256 + (size_t)B * 4); // B u8

    size_t need = 256 + (size_t)B * 5;
    int store = (ws_size >= need) ? 1 : 0;  // spill H2/err, else recompute in pass2

    uce_init<<<1, 32, 0, stream>>>(minmax, gcount, gsh, gse);

    const int threads = 256;
    int blocks = (B + threads - 1) / threads;
    if (blocks > 2048) blocks = 2048;
    if (blocks < 1) blocks = 1;

    uce_pass1<<<blocks, threads, 0, stream>>>(
        logits, labels, minmax, store ? h2arr : (float*)nullptr,
        store ? errarr : (unsigned char*)nullptr, B, store);

    uce_pass2<<<blocks, threads, 0, stream>>>(
        h2arr, errarr, logits, labels, minmax, gcount, gsh, gse, B, store);

    uce_final<<<1, 32, 0, stream>>>(gcount, gsh, gse, out, B);
}